// SSIMLoss_60997125537893
// MI455X (gfx1250) — compile-verified
//
#include <hip/hip_runtime.h>
#include <math.h>

// ---------------------------------------------------------------------------
// SSIM loss for (16,3,512,512) fp32, skimage-compatible (7x7 box, VALID crop).
// HBM-bound problem (~100MB in, floor ~4.3us @ 23.3TB/s). Horizontal 7-tap box
// sums are done on the CDNA5 matrix pipe with V_WMMA_F32_16X16X4_F32 against a
// 0/1 banded matrix (exact in f32); vertical 7-tap + SSIM pointwise on VALU.
// ---------------------------------------------------------------------------

typedef __attribute__((ext_vector_type(2))) float v2f;
typedef __attribute__((ext_vector_type(8))) float v8f;

#define NS      16
#define CH      3
#define IMG     512
#define OUT_DIM 506          // 512 - 6
#define TPD     32           // tiles per dim: ceil(506/16)
#define TILES_PER_SAMPLE (CH * TPD * TPD)       // 3072
#define TOTAL_TILES      (NS * TILES_PER_SAMPLE) // 49152
#define WAVES_PER_BLOCK  4
#define NBLOCKS3         (TOTAL_TILES / WAVES_PER_BLOCK) // 12288

// per-wave LDS layout (floats), stride-24 rows:
//   [0,528)        : p tile   22 rows x 24 cols
//   [528,1056)     : t tile   22 rows x 24 cols
//   [1056,1056+5*384) : 5 colsum buffers, 16 rows x 24 cols each
#define LDS_PER_WAVE 2976
#define CS_OFF       1056

// ---- kernel 1: per-sample partial min/max over pred -----------------------
__global__ void ssim_minmax_part(const float* __restrict__ pred,
                                 float* __restrict__ pmin,
                                 float* __restrict__ pmax) {
  __shared__ float smn[256], smx[256];
  const int s = blockIdx.y;        // sample
  const int b = blockIdx.x;        // chunk 0..15
  const float* base = pred + (size_t)s * (CH * IMG * IMG);
  const int chunk = (CH * IMG * IMG) / 16;     // 49152
  float mn = 3.402823466e38f, mx = -3.402823466e38f;
  for (int i = b * chunk + threadIdx.x; i < (b + 1) * chunk; i += 256) {
    float v = base[i];
    mn = fminf(mn, v);
    mx = fmaxf(mx, v);
  }
  smn[threadIdx.x] = mn; smx[threadIdx.x] = mx;
  __syncthreads();
  for (int st = 128; st > 0; st >>= 1) {
    if (threadIdx.x < st) {
      smn[threadIdx.x] = fminf(smn[threadIdx.x], smn[threadIdx.x + st]);
      smx[threadIdx.x] = fmaxf(smx[threadIdx.x], smx[threadIdx.x + st]);
    }
    __syncthreads();
  }
  if (threadIdx.x == 0) {
    pmin[s * 16 + b] = smn[0];
    pmax[s * 16 + b] = smx[0];
  }
}

// ---- kernel 2: data_range -> C1, C2 per sample ----------------------------
__global__ void ssim_minmax_fin(const float* __restrict__ pmin,
                                const float* __restrict__ pmax,
                                float* __restrict__ c1c2) {
  const int s = blockIdx.x;
  if (threadIdx.x == 0) {
    float mn = pmin[s * 16], mx = pmax[s * 16];
    for (int i = 1; i < 16; ++i) {
      mn = fminf(mn, pmin[s * 16 + i]);
      mx = fmaxf(mx, pmax[s * 16 + i]);
    }
    float dr = mx - mn;
    float k1dr = 0.01f * dr, k2dr = 0.03f * dr;
    c1c2[s]      = k1dr * k1dr;   // C1
    c1c2[16 + s] = k2dr * k2dr;   // C2
  }
}

// ---- kernel 3: per-tile SSIM sums; one wave32 handles one 16x16 out tile --
__global__ void __launch_bounds__(128) ssim_tiles(
    const float* __restrict__ pred, const float* __restrict__ targ,
    const float* __restrict__ c1c2, float* __restrict__ partials) {
  __shared__ float lds[WAVES_PER_BLOCK][LDS_PER_WAVE];
  __shared__ float red[128];
  const int tid  = threadIdx.x;
  const int wave = tid >> 5;
  const int lane = tid & 31;
  float* W = lds[wave];

  // tile decomposition
  const int tix  = blockIdx.x * WAVES_PER_BLOCK + wave;     // 0..49151
  const int sIdx = tix / TILES_PER_SAMPLE;
  const int rem  = tix - sIdx * TILES_PER_SAMPLE;
  const int cIdx = rem / (TPD * TPD);
  const int t2   = rem - cIdx * (TPD * TPD);
  const int ty = t2 >> 5, tx = t2 & 31;
  const int y0 = ty * 16, x0 = tx * 16;

  const size_t plane = (size_t)(sIdx * CH + cIdx) * (IMG * IMG);
  const float* P = pred + plane;
  const float* T = targ + plane;

  // ---- phase 1: load 22x22 halo tiles of p and t into LDS (stride 24) ----
  for (int idx = lane; idx < 22 * 24; idx += 32) {
    int r  = idx / 24;
    int cc = idx - r * 24;
    float pv = 0.0f, tv = 0.0f;
    if (cc < 22) {
      int y = y0 + r;  y = (y > IMG - 1) ? IMG - 1 : y;   // clamped rows/cols
      int x = x0 + cc; x = (x > IMG - 1) ? IMG - 1 : x;   // only feed masked outs
      size_t off = (size_t)y * IMG + x;
      pv = P[off];
      tv = T[off];
    }
    W[idx]       = pv;
    W[528 + idx] = tv;
  }
  __syncthreads();

  // ---- phase 2: vertical 7-tap sums for 5 quantities (16 x 22, pad to 24) --
  for (int idx = lane; idx < 16 * 24; idx += 32) {
    int r  = idx / 24;
    int cc = idx - r * 24;
    float sp = 0.f, st = 0.f, spp = 0.f, stt = 0.f, spt = 0.f;
    if (cc < 22) {
#pragma unroll
      for (int dy = 0; dy < 7; ++dy) {
        float pv = W[(r + dy) * 24 + cc];
        float tv = W[528 + (r + dy) * 24 + cc];
        sp += pv;
        st += tv;
        spp = fmaf(pv, pv, spp);
        stt = fmaf(tv, tv, stt);
        spt = fmaf(pv, tv, spt);
      }
    }
    W[CS_OFF + 0 * 384 + idx] = sp;
    W[CS_OFF + 1 * 384 + idx] = st;
    W[CS_OFF + 2 * 384 + idx] = spp;
    W[CS_OFF + 3 * 384 + idx] = stt;
    W[CS_OFF + 4 * 384 + idx] = spt;
  }
  __syncthreads();

  // ---- phase 3: horizontal 7-tap box sum via WMMA f32 16x16x4 -------------
  // Z(16x16) = Y(16 x 22) * N(22 x 16), N[K][j] = 1 iff j <= K <= j+6.
  // A layout (16x4 f32): lane holds row (lane&15); VGPR0/1 hold K = 4k + 2*(lane>>4) + {0,1}.
  // B layout (4x16 f32): VGPR v holds row K = 4k + v + 2*(lane>>4), col = lane&15.
  const int half = lane >> 4;
  const int mrow = lane & 15;
  const float* cs = &W[CS_OFF];
  v8f acc0 = {}, acc1 = {}, acc2 = {}, acc3 = {}, acc4 = {};
#pragma unroll
  for (int k = 0; k < 6; ++k) {
    const int K0 = 4 * k + 2 * half;
    v2f b;
    b.x = (K0 >= mrow && K0 - mrow <= 6) ? 1.0f : 0.0f;
    b.y = ((K0 + 1) >= mrow && (K0 + 1) - mrow <= 6) ? 1.0f : 0.0f;
    v2f a;
    a.x = cs[0 * 384 + mrow * 24 + K0];
    a.y = cs[0 * 384 + mrow * 24 + K0 + 1];
    acc0 = __builtin_amdgcn_wmma_f32_16x16x4_f32(false, a, false, b, (short)0, acc0, false, false);
    a.x = cs[1 * 384 + mrow * 24 + K0];
    a.y = cs[1 * 384 + mrow * 24 + K0 + 1];
    acc1 = __builtin_amdgcn_wmma_f32_16x16x4_f32(false, a, false, b, (short)0, acc1, false, false);
    a.x = cs[2 * 384 + mrow * 24 + K0];
    a.y = cs[2 * 384 + mrow * 24 + K0 + 1];
    acc2 = __builtin_amdgcn_wmma_f32_16x16x4_f32(false, a, false, b, (short)0, acc2, false, false);
    a.x = cs[3 * 384 + mrow * 24 + K0];
    a.y = cs[3 * 384 + mrow * 24 + K0 + 1];
    acc3 = __builtin_amdgcn_wmma_f32_16x16x4_f32(false, a, false, b, (short)0, acc3, false, false);
    a.x = cs[4 * 384 + mrow * 24 + K0];
    a.y = cs[4 * 384 + mrow * 24 + K0 + 1];
    acc4 = __builtin_amdgcn_wmma_f32_16x16x4_f32(false, a, false, b, (short)0, acc4, false, false);
  }

  // ---- phase 4: pointwise SSIM on the 8 D-elements this lane owns ---------
  const float C1 = c1c2[sIdx];
  const float C2 = c1c2[16 + sIdx];
  float lsum = 0.0f;
  const int ox = x0 + mrow;
#pragma unroll
  for (int r = 0; r < 8; ++r) {
    int oy = y0 + r + 8 * half;     // D: vgpr r -> row r + 8*(lane>>4)
    if (oy < OUT_DIM && ox < OUT_DIM) {
      float sp  = acc0[r], stq = acc1[r];
      float spp = acc2[r], stt = acc3[r], spt = acc4[r];
      float ux = sp  * (1.0f / 49.0f);
      float uy = stq * (1.0f / 49.0f);
      float vx  = (spp - sp  * sp  * (1.0f / 49.0f)) * (1.0f / 48.0f);
      float vy  = (stt - stq * stq * (1.0f / 49.0f)) * (1.0f / 48.0f);
      float vxy = (spt - sp  * stq * (1.0f / 49.0f)) * (1.0f / 48.0f);
      float num = (2.0f * ux * uy + C1) * (2.0f * vxy + C2);
      float den = (ux * ux + uy * uy + C1) * (vx + vy + C2);
      lsum += num / den;
    }
  }

  // ---- block reduction: 4 tiles -> 1 partial ------------------------------
  __syncthreads();
  red[tid] = lsum;
  __syncthreads();
  for (int st = 64; st > 0; st >>= 1) {
    if (tid < st) red[tid] += red[tid + st];
    __syncthreads();
  }
  if (tid == 0) partials[blockIdx.x] = red[0];
}

// ---- kernel 4: final reduction -> 1 - mean(SSIM) --------------------------
__global__ void ssim_final(const float* __restrict__ partials,
                           float* __restrict__ out) {
  __shared__ float red[256];
  float sum = 0.0f;
  for (int i = threadIdx.x; i < NBLOCKS3; i += 256) sum += partials[i];
  red[threadIdx.x] = sum;
  __syncthreads();
  for (int st = 128; st > 0; st >>= 1) {
    if (threadIdx.x < st) red[threadIdx.x] += red[threadIdx.x + st];
    __syncthreads();
  }
  if (threadIdx.x == 0) {
    out[0] = 1.0f - red[0] * (1.0f /
        ((float)NS * (float)CH * (float)OUT_DIM * (float)OUT_DIM));
  }
}

// ---------------------------------------------------------------------------
extern "C" void kernel_launch(void* const* d_in, const int* in_sizes, int n_in,
                              void* d_out, int out_size, void* d_ws, size_t ws_size,
                              hipStream_t stream) {
  (void)in_sizes; (void)n_in; (void)out_size; (void)ws_size;
  const float* pred = (const float*)d_in[0];
  const float* targ = (const float*)d_in[1];
  float* out = (float*)d_out;
  float* ws  = (float*)d_ws;

  // ws layout (floats): [0,256) pmin, [256,512) pmax, [512,544) C1|C2,
  //                     [1024, 1024+12288) tile partials   (~53 KB total)
  float* pmin     = ws;
  float* pmax     = ws + 256;
  float* c1c2     = ws + 512;
  float* partials = ws + 1024;

  ssim_minmax_part<<<dim3(16, NS), 256, 0, stream>>>(pred, pmin, pmax);
  ssim_minmax_fin<<<NS, 32, 0, stream>>>(pmin, pmax, c1c2);
  ssim_tiles<<<NBLOCKS3, 128, 0, stream>>>(pred, targ, c1c2, partials);
  ssim_final<<<1, 256, 0, stream>>>(partials, out);
}